// MemoryAugmentedNetwork_84799834293133
// MI455X (gfx1250) — compile-verified
//
#include <hip/hip_runtime.h>
#include <hip/hip_bf16.h>
#include <math.h>

typedef __attribute__((ext_vector_type(2))) float v2f;
typedef __attribute__((ext_vector_type(4))) float v4f;
typedef __attribute__((ext_vector_type(8))) float v8f;

#define B_   2048
#define IN_  1024
#define E_   512
#define M_   256
#define NB_  8
#define H_   8
#define D_   64

// ---------------------------------------------------------------------------
// C[M,N] = act(A[M,K] @ W[N,K]^T + bias[N])   (bias may be null, ACT=1 -> tanh)
// Register-blocked: each wave computes a 32x64 macro-tile = 2x4 WMMA tiles,
// so per k-step we load 2 A fragments + 4 B fragments (6 x b64) and issue
// 8 x V_WMMA_F32_16X16X4_F32 (2.7x fewer bytes/FLOP than 1 tile/wave).
// f32 A 16x4 layout: lanes 0-15 {K=0,K=1}, lanes 16-31 {K=2,K=3}, M=lane&15.
// f32 B 4x16 layout mirrors A with N=lane&15. C/D: VGPR r -> M=r (+8 hi half).
// Grid: (N/64, M/128), 128 threads (4 waves, each wave = 2 m-tiles).
// ---------------------------------------------------------------------------
template<int ACT>
__global__ __launch_bounds__(128)
void gemm_xwT(const float* __restrict__ A, const float* __restrict__ W,
              const float* __restrict__ bias, float* __restrict__ C,
              int M, int N, int K)
{
    const int lane = threadIdx.x & 31;
    const int wv   = threadIdx.x >> 5;
    const int ln15 = lane & 15;
    const int koff = (lane >> 4) << 1;          // 0 or 2

    const int tm0 = (blockIdx.y * 4 + wv) * 2;  // two 16-row tiles
    const int tn0 = blockIdx.x * 4;             // four 16-col tiles

    const float* a0 = A + (size_t)(tm0 * 16 + ln15) * K + koff;
    const float* a1 = a0 + (size_t)16 * K;
    const float* w0 = W + (size_t)(tn0 * 16 + ln15) * K + koff;
    const float* w1 = w0 + (size_t)16 * K;
    const float* w2 = w0 + (size_t)32 * K;
    const float* w3 = w0 + (size_t)48 * K;

    v8f acc[2][4];
    #pragma unroll
    for (int i = 0; i < 2; ++i)
        #pragma unroll
        for (int j = 0; j < 4; ++j)
            acc[i][j] = (v8f){0.f, 0.f, 0.f, 0.f, 0.f, 0.f, 0.f, 0.f};

    for (int k = 0; k < K; k += 4) {
        const v2f av0 = *(const v2f*)(a0 + k);
        const v2f av1 = *(const v2f*)(a1 + k);
        const v2f bv0 = *(const v2f*)(w0 + k);
        const v2f bv1 = *(const v2f*)(w1 + k);
        const v2f bv2 = *(const v2f*)(w2 + k);
        const v2f bv3 = *(const v2f*)(w3 + k);
        acc[0][0] = __builtin_amdgcn_wmma_f32_16x16x4_f32(false, av0, false, bv0, (short)0, acc[0][0], false, false);
        acc[0][1] = __builtin_amdgcn_wmma_f32_16x16x4_f32(false, av0, false, bv1, (short)0, acc[0][1], false, false);
        acc[0][2] = __builtin_amdgcn_wmma_f32_16x16x4_f32(false, av0, false, bv2, (short)0, acc[0][2], false, false);
        acc[0][3] = __builtin_amdgcn_wmma_f32_16x16x4_f32(false, av0, false, bv3, (short)0, acc[0][3], false, false);
        acc[1][0] = __builtin_amdgcn_wmma_f32_16x16x4_f32(false, av1, false, bv0, (short)0, acc[1][0], false, false);
        acc[1][1] = __builtin_amdgcn_wmma_f32_16x16x4_f32(false, av1, false, bv1, (short)0, acc[1][1], false, false);
        acc[1][2] = __builtin_amdgcn_wmma_f32_16x16x4_f32(false, av1, false, bv2, (short)0, acc[1][2], false, false);
        acc[1][3] = __builtin_amdgcn_wmma_f32_16x16x4_f32(false, av1, false, bv3, (short)0, acc[1][3], false, false);
    }

    const int cm0 = tm0 * 16 + ((lane >> 4) << 3);
    #pragma unroll
    for (int j = 0; j < 4; ++j) {
        const int   cn = (tn0 + j) * 16 + ln15;
        const float bv = bias ? bias[cn] : 0.f;
        #pragma unroll
        for (int r = 0; r < 8; ++r) {
            float v0 = acc[0][j][r] + bv;
            float v1 = acc[1][j][r] + bv;
            if (ACT == 1) { v0 = tanhf(v0); v1 = tanhf(v1); }
            C[(size_t)(cm0 + r) * N + cn]      = v0;
            C[(size_t)(cm0 + 16 + r) * N + cn] = v1;
        }
    }
}

// ---------------------------------------------------------------------------
// Attention 1: q = memory_updates (B,8,E), k=v = broadcast memory -> K0,V0.
// One block per (b,h); 256 threads. scores (8q x 256l), softmax over l,
// ctx = P @ V0_h.  QP1/CTX1 layout: b*4096 + q*512 + h*64 + d.
// ---------------------------------------------------------------------------
__global__ __launch_bounds__(256)
void attn1_kernel(const float* __restrict__ QP1, const float* __restrict__ K0,
                  const float* __restrict__ V0, float* __restrict__ CTX1)
{
    __shared__ float qs[NB_][D_];   // 2 KB
    __shared__ float P[NB_][M_];    // 8 KB
    const int b = blockIdx.x;
    const int h = blockIdx.y;
    const int t = threadIdx.x;

    for (int i = t; i < NB_ * D_; i += 256) {
        const int q = i >> 6, d = i & 63;
        qs[q][d] = QP1[(size_t)b * 4096 + (size_t)q * E_ + h * D_ + d];
    }
    __syncthreads();

    {   // scores: thread t <-> key row l = t
        const int l = t;
        const float* k0 = K0 + (size_t)l * E_ + h * D_;
        float acc[8] = {0.f, 0.f, 0.f, 0.f, 0.f, 0.f, 0.f, 0.f};
        for (int d = 0; d < D_; d += 4) {
            const v4f kk = *(const v4f*)(k0 + d);
            #pragma unroll
            for (int q = 0; q < 8; ++q)
                acc[q] += qs[q][d] * kk[0] + qs[q][d + 1] * kk[1]
                        + qs[q][d + 2] * kk[2] + qs[q][d + 3] * kk[3];
        }
        #pragma unroll
        for (int q = 0; q < 8; ++q) P[q][l] = acc[q] * 0.125f; // 1/sqrt(64)
    }
    __syncthreads();

    {   // softmax over l: wave g handles q = g
        const int q = t >> 5, lane = t & 31;
        float e[8];
        float mx = -INFINITY;
        for (int i = 0; i < 8; ++i) mx = fmaxf(mx, P[q][lane + 32 * i]);
        for (int off = 16; off; off >>= 1) mx = fmaxf(mx, __shfl_xor(mx, off, 32));
        float sum = 0.f;
        for (int i = 0; i < 8; ++i) { e[i] = __expf(P[q][lane + 32 * i] - mx); sum += e[i]; }
        for (int off = 16; off; off >>= 1) sum += __shfl_xor(sum, off, 32);
        const float inv = 1.f / sum;
        for (int i = 0; i < 8; ++i) P[q][lane + 32 * i] = e[i] * inv;
    }
    __syncthreads();

    {   // ctx[q][d] = sum_l P[q][l] * V0[l,h,d];  d = lane, lane+32
        const int q = t >> 5, lane = t & 31;
        float c0 = 0.f, c1 = 0.f;
        for (int l = 0; l < M_; ++l) {
            const float p = P[q][l];
            const float* v0 = V0 + (size_t)l * E_ + h * D_;
            c0 += p * v0[lane];
            c1 += p * v0[lane + 32];
        }
        const size_t o = (size_t)b * 4096 + (size_t)q * E_ + h * D_;
        CTX1[o + lane]      = c0;
        CTX1[o + lane + 32] = c1;
    }
}

// ---------------------------------------------------------------------------
// S[b,e] = mean_q CTX1[b,q,e]
// ---------------------------------------------------------------------------
__global__ __launch_bounds__(256)
void mean_q_kernel(const float* __restrict__ CTX1, float* __restrict__ S)
{
    const size_t i = (size_t)blockIdx.x * 256 + threadIdx.x;  // < B*E
    const size_t b = i >> 9, e = i & 511;
    const float* p = CTX1 + b * 4096 + e;
    float a = 0.f;
    #pragma unroll
    for (int q = 0; q < 8; ++q) a += p[q * 512];
    S[i] = a * 0.125f;
}

// ---------------------------------------------------------------------------
// updated_memory[b,l,:] = memory[l,:] + u[b,:]   (1.07 GB stream, NT stores)
// ---------------------------------------------------------------------------
__global__ __launch_bounds__(256)
void upd_mem_kernel(const float* __restrict__ MEMORY, const float* __restrict__ U,
                    float* __restrict__ OUT)
{
    const size_t idx = (size_t)blockIdx.x * 256 + threadIdx.x; // float4 index
    const size_t e4 = idx & 127;          // E/4
    const size_t l  = (idx >> 7) & 255;   // M
    const size_t b  = idx >> 15;
    const v4f mv = ((const v4f*)MEMORY)[l * 128 + e4];
    const v4f uv = ((const v4f*)U)[b * 128 + e4];
    const v4f r  = mv + uv;
    __builtin_nontemporal_store(r, ((v4f*)OUT) + idx);
}

// ---------------------------------------------------------------------------
// Attention 2: q = encoded (B,1,E).  kp = K0 + uk (uk shifts all keys equally
// -> cancels in softmax);  ctx = attn @ V0 + uv  (since attn rows sum to 1).
// One block per b; wave h handles head h.
// ---------------------------------------------------------------------------
__global__ __launch_bounds__(256)
void attn2_kernel(const float* __restrict__ QP2, const float* __restrict__ K0,
                  const float* __restrict__ V0, const float* __restrict__ UV,
                  float* __restrict__ CTX2)
{
    __shared__ float qs[H_][D_];
    __shared__ float P[H_][M_];
    const int b = blockIdx.x;
    const int t = threadIdx.x;
    const int h = t >> 5;
    const int lane = t & 31;

    for (int i = t; i < E_; i += 256) qs[i >> 6][i & 63] = QP2[(size_t)b * E_ + i];
    __syncthreads();

    float sc[8];
    float mx = -INFINITY;
    for (int i = 0; i < 8; ++i) {
        const int l = lane + 32 * i;
        const float* k0 = K0 + (size_t)l * E_ + h * D_;
        float acc = 0.f;
        #pragma unroll
        for (int d = 0; d < D_; d += 4) {
            const v4f kk = *(const v4f*)(k0 + d);
            acc += qs[h][d] * kk[0] + qs[h][d + 1] * kk[1]
                 + qs[h][d + 2] * kk[2] + qs[h][d + 3] * kk[3];
        }
        sc[i] = acc * 0.125f;
        mx = fmaxf(mx, sc[i]);
    }
    for (int off = 16; off; off >>= 1) mx = fmaxf(mx, __shfl_xor(mx, off, 32));
    float sum = 0.f;
    for (int i = 0; i < 8; ++i) { sc[i] = __expf(sc[i] - mx); sum += sc[i]; }
    for (int off = 16; off; off >>= 1) sum += __shfl_xor(sum, off, 32);
    const float inv = 1.f / sum;
    for (int i = 0; i < 8; ++i) P[h][lane + 32 * i] = sc[i] * inv;
    __syncthreads();

    float c0 = 0.f, c1 = 0.f;
    for (int l = 0; l < M_; ++l) {
        const float p = P[h][l];
        const float* v0 = V0 + (size_t)l * E_ + h * D_;
        c0 += p * v0[lane];
        c1 += p * v0[lane + 32];
    }
    const size_t o = (size_t)b * E_ + h * D_;
    CTX2[o + lane]      = c0 + UV[o + lane];
    CTX2[o + lane + 32] = c1 + UV[o + lane + 32];
}

// ---------------------------------------------------------------------------
static inline void launch_gemm(const float* A, const float* W, const float* bias,
                               float* C, int M, int N, int K, int act,
                               hipStream_t s)
{
    dim3 grid(N / 64, M / 128), blk(128);
    if (act)
        hipLaunchKernelGGL((gemm_xwT<1>), grid, blk, 0, s, A, W, bias, C, M, N, K);
    else
        hipLaunchKernelGGL((gemm_xwT<0>), grid, blk, 0, s, A, W, bias, C, M, N, K);
}

extern "C" void kernel_launch(void* const* d_in, const int* in_sizes, int n_in,
                              void* d_out, int out_size, void* d_ws, size_t ws_size,
                              hipStream_t stream)
{
    const float* x      = (const float*)d_in[0];   // (2048,1024)
    const float* memory = (const float*)d_in[1];   // (256,512)
    const float* enc_w  = (const float*)d_in[2];   // (512,1024)
    const float* enc_b  = (const float*)d_in[3];   // (512)
    const float* blk_w  = (const float*)d_in[4];   // (8,512,512) -> (4096,512)
    const float* blk_b  = (const float*)d_in[5];   // (8,512) -> (4096)
    const float* in_w   = (const float*)d_in[6];   // (1536,512)
    const float* in_b   = (const float*)d_in[7];   // (1536)
    const float* out_w  = (const float*)d_in[8];   // (512,512)
    const float* out_b  = (const float*)d_in[9];   // (512)
    const float* dec_w  = (const float*)d_in[10];  // (1024,512)
    const float* dec_b  = (const float*)d_in[11];  // (1024)

    const float* wq = in_w;
    const float* wk = in_w + (size_t)E_ * E_;
    const float* wv = in_w + (size_t)2 * E_ * E_;
    const float* bq = in_b;
    const float* bk = in_b + E_;
    const float* bv = in_b + 2 * E_;

    float* out     = (float*)d_out;                 // output (2048,1024)
    float* out_mem = out + (size_t)B_ * IN_;        // updated_memory (2048,256,512)

    char* ws = (char*)d_ws;
    const size_t MiB = 1024 * 1024;
    float* ENC  = (float*)(ws + 0 * MiB);    // (2048,512)    4 MiB
    float* MU   = (float*)(ws + 4 * MiB);    // (2048,4096)  32 MiB (reused as CTX1)
    float* QP1  = (float*)(ws + 36 * MiB);   // (16384,512)  32 MiB
    float* K0   = (float*)(ws + 68 * MiB);   // (256,512)   0.5 MiB
    float* V0   = (float*)(ws + 68 * MiB + 512 * 1024);
    float* S    = (float*)(ws + 69 * MiB);   // (2048,512)
    float* U    = (float*)(ws + 73 * MiB);   // (2048,512)
    float* QP2  = (float*)(ws + 77 * MiB);   // (2048,512)
    float* UV   = (float*)(ws + 81 * MiB);   // (2048,512)
    float* CTX2 = (float*)(ws + 85 * MiB);   // (2048,512)
    float* RET  = (float*)(ws + 89 * MiB);   // (2048,512)
    float* CTX1 = MU;                        // MU dead after QP1 is built

    // 1. encoded = tanh(x @ enc_w.T + enc_b)
    launch_gemm(x, enc_w, enc_b, ENC, B_, E_, IN_, 1, stream);
    // 2. memory_updates = encoded @ blk_w.T + blk_b    (N = 8*512)
    launch_gemm(ENC, blk_w, blk_b, MU, B_, NB_ * E_, E_, 0, stream);
    // 3. qp1 = memory_updates @ wq.T + bq              (M = B*Lq)
    launch_gemm(MU, wq, bq, QP1, B_ * NB_, E_, E_, 0, stream);
    // 4. K0 = memory @ wk.T + bk ; V0 = memory @ wv.T + bv   (batch-invariant)
    launch_gemm(memory, wk, bk, K0, M_, E_, E_, 0, stream);
    launch_gemm(memory, wv, bv, V0, M_, E_, E_, 0, stream);
    // 5. first attention
    {
        dim3 grid(B_, H_), blk(256);
        hipLaunchKernelGGL(attn1_kernel, grid, blk, 0, stream, QP1, K0, V0, CTX1);
    }
    // 6. s = mean_q ctx1 ; u = s @ out_w.T + out_b
    hipLaunchKernelGGL(mean_q_kernel, dim3((B_ * E_) / 256), dim3(256), 0, stream,
                       CTX1, S);
    launch_gemm(S, out_w, out_b, U, B_, E_, E_, 0, stream);
    // 7. updated_memory = memory + u  (output #2, 1.07 GB nontemporal stream)
    hipLaunchKernelGGL(upd_mem_kernel,
                       dim3(((size_t)B_ * M_ * E_ / 4) / 256), dim3(256), 0, stream,
                       memory, U, out_mem);
    // 8. qp2 = encoded @ wq.T + bq ; uv = u @ wv.T (uk cancels in softmax)
    launch_gemm(ENC, wq, bq, QP2, B_, E_, E_, 0, stream);
    launch_gemm(U, wv, nullptr, UV, B_, E_, E_, 0, stream);
    // 9. second attention (retrieved context)
    hipLaunchKernelGGL(attn2_kernel, dim3(B_), dim3(256), 0, stream,
                       QP2, K0, V0, UV, CTX2);
    // 10. retrieved = ctx2 @ out_w.T + out_b ; output = retrieved @ dec_w.T + dec_b
    launch_gemm(CTX2, out_w, out_b, RET, B_, E_, E_, 0, stream);
    launch_gemm(RET, dec_w, dec_b, out, B_, IN_, E_, 0, stream);
}